// GAT_50225347560039
// MI455X (gfx1250) — compile-verified
//
#include <hip/hip_runtime.h>
#include <hip/hip_bf16.h>
#include <math.h>

typedef __attribute__((ext_vector_type(2))) float v2f;
typedef __attribute__((ext_vector_type(8))) float v8f;

#define FIN   256   // input features
#define FMID  256   // H1*hidden
#define H1N   4
#define NC    40    // nclasses
#define NCP   48    // padded nclasses (3 x 16 col tiles)
#define PB    128   // batchnorm partial blocks
#define NEG_SLOPE 0.2f
#define SM_EPS 1e-16f
#define BN_EPS 1e-5f

// ---- order-preserving float<->uint encoding for atomicMax on floats ----
__device__ __forceinline__ unsigned encf(float f) {
  unsigned u = __float_as_uint(f);
  return (u & 0x80000000u) ? ~u : (u | 0x80000000u);
}
__device__ __forceinline__ float decf(unsigned u) {
  return (u & 0x80000000u) ? __uint_as_float(u & 0x7FFFFFFFu) : __uint_as_float(~u);
}
__device__ __forceinline__ float lrelu(float v) { return v > 0.f ? v : NEG_SLOPE * v; }

// ---------------------------------------------------------------- zero fill
__global__ void zero_u32(unsigned* __restrict__ p, size_t n) {
  size_t i = (size_t)blockIdx.x * blockDim.x + threadIdx.x;
  size_t stride = (size_t)gridDim.x * blockDim.x;
  for (; i < n; i += stride) p[i] = 0u;
}

// ---------------------------------------------------------------- weight pack
// Pack W[K,Mvalid] (row-major) into WMMA-B fragment pairs, zero-padding to M
// columns. Pair index (kk*2 + half)*M + col holds {W[4kk+2half][col],
// W[4kk+2half+1][col]} so a lane's B fragment is ONE b64 load; the per-K-step
// pointer advance is a constant (+4*M floats).
__global__ void packW(const float* __restrict__ W, float* __restrict__ Wp,
                      int K, int M, int Mvalid) {
  int i = blockIdx.x * blockDim.x + threadIdx.x;     // pair index
  int total = (K >> 2) * 2 * M;
  if (i >= total) return;
  int col = i % M;
  int t = i / M;                 // kk*2 + half
  int k = (t >> 1) * 4 + (t & 1) * 2;
  float x0 = 0.f, x1 = 0.f;
  if (col < Mvalid) {
    x0 = W[(size_t)k * Mvalid + col];
    x1 = W[(size_t)(k + 1) * Mvalid + col];
  }
  Wp[(size_t)i * 2]     = x0;
  Wp[(size_t)i * 2 + 1] = x1;
}

// ------------------------------------------------------- f32 WMMA GEMM strip
// Y[N,M] = X[N,K] @ W[K,Mvalid(padded to M)]; one wave owns a 16x(16*CT) strip.
// No guards: N%16==0, M == ld == 16*CT*nColGroups, K%4==0. Wp is packW output.
template <int CT>
__global__ void gemm_f32_wmma(const float* __restrict__ X, const float* __restrict__ Wp,
                              float* __restrict__ Y, int nRowTiles, int nColGroups,
                              int K, int M) {
  int gw   = (int)((blockIdx.x * blockDim.x + threadIdx.x) >> 5);
  int lane = threadIdx.x & 31;
  if (gw >= nRowTiles * nColGroups) return;          // wave-uniform guard
  int rowTile = gw / nColGroups, colGroup = gw % nColGroups;
  int row0 = rowTile << 4, col0 = colGroup * (CT << 4);
  int mn  = lane & 15;                // A: row in tile / B: col in tile
  int kbh = lane >> 4;                // K-pair half: lanes<16 -> K 0,1 ; >=16 -> K 2,3
  const float* xr = X + (size_t)(row0 + mn) * K + (kbh << 1);
  const float* bp = Wp + ((size_t)kbh * M + col0 + mn) * 2;
  v8f acc[CT];
#pragma unroll
  for (int t = 0; t < CT; ++t) acc[t] = (v8f){};
  for (int kk = 0; kk < (K >> 2); ++kk) {
    v2f a = *(const v2f*)xr;
    v2f b[CT];
#pragma unroll
    for (int t = 0; t < CT; ++t) b[t] = *(const v2f*)(bp + t * 32);  // 16 cols * 2
#pragma unroll
    for (int t = 0; t < CT; ++t)
      acc[t] = __builtin_amdgcn_wmma_f32_16x16x4_f32(false, a, false, b[t], (short)0,
                                                     acc[t], false, false);
    xr += 4;
    bp += (size_t)4 * M;            // next K-step: +2M pairs
  }
  int rbase = row0 + (kbh << 3);    // C/D: row = vgpr + 8*(lane>>4)
#pragma unroll
  for (int t = 0; t < CT; ++t)
#pragma unroll
    for (int r = 0; r < 8; ++r)
      Y[(size_t)(rbase + r) * M + col0 + t * 16 + mn] = acc[t][r];
}

// ------------------------------------------- per-node attention scores, layer1
__global__ void scores1(const float* __restrict__ h1, const float* __restrict__ asrc,
                        const float* __restrict__ adst, float* __restrict__ s,
                        float* __restrict__ d, int N) {
  __shared__ float ps[FMID], pd[FMID];
  int n = blockIdx.x, t = threadIdx.x;
  float v = h1[(size_t)n * FMID + t];
  ps[t] = v * asrc[t];
  pd[t] = v * adst[t];
  __syncthreads();
  for (int off = 32; off >= 1; off >>= 1) {
    if ((t & 63) < off) { ps[t] += ps[t + off]; pd[t] += pd[t + off]; }
    __syncthreads();
  }
  if ((t & 63) == 0) {
    int h = t >> 6;
    s[(size_t)n * H1N + h] = ps[t];
    d[(size_t)n * H1N + h] = pd[t];
  }
}

// ----------------------------------------------- layer1 edge pass A: segment max
__global__ void edgeA1(const long long* __restrict__ ei, const float* __restrict__ s,
                       const float* __restrict__ d, unsigned* __restrict__ m,
                       int E, int N) {
  int idx = blockIdx.x * blockDim.x + threadIdx.x;
  int tot = (E + N) * H1N;
  if (idx >= tot) return;
  int e = idx >> 2, h = idx & 3;
  int src, dst;
  if (e < E) { src = (int)ei[e]; dst = (int)ei[(size_t)E + e]; }
  else       { src = dst = e - E; }
  float v = lrelu(s[(size_t)src * H1N + h] + d[(size_t)dst * H1N + h]);
  atomicMax(&m[(size_t)dst * H1N + h], encf(v));
}

// --------------------------- layer1 edge pass B: exp, denom, weighted scatter-sum
__global__ void edgeB1(const long long* __restrict__ ei, const float* __restrict__ s,
                       const float* __restrict__ d, const unsigned* __restrict__ m,
                       float* __restrict__ den, const float* __restrict__ h1,
                       float* __restrict__ agg, int E, int N) {
  int gw = (int)((blockIdx.x * blockDim.x + threadIdx.x) >> 5);
  int lane = threadIdx.x & 31;
  if (gw >= E + N) return;
  int src, dst;
  if (gw < E) { src = (int)ei[gw]; dst = (int)ei[(size_t)E + gw]; }
  else        { src = dst = gw - E; }
  float p[H1N];
#pragma unroll
  for (int h = 0; h < H1N; ++h) {
    float v = lrelu(s[(size_t)src * H1N + h] + d[(size_t)dst * H1N + h]);
    p[h] = expf(v - decf(m[(size_t)dst * H1N + h]));
  }
  if (lane < H1N) atomicAdd(&den[(size_t)dst * H1N + lane], p[lane]);
  const float* hs = h1 + (size_t)src * FMID;
  float* ad = agg + (size_t)dst * FMID;
#pragma unroll
  for (int i = 0; i < 8; ++i) {
    int ch = lane + 32 * i;        // consecutive lanes -> coalesced
    atomicAdd(&ad[ch], hs[ch] * p[i >> 1]);
  }
}

// ------- normalize by denom, add bias, accumulate deterministic BN partials
__global__ void bnstats(float* __restrict__ agg, const float* __restrict__ den,
                        const float* __restrict__ b1, float* __restrict__ part, int N) {
  int c = threadIdx.x, bb = blockIdx.x;  // bb in [0, PB)
  float sum = 0.f, sq = 0.f;
  for (int n = bb; n < N; n += PB) {
    size_t i = (size_t)n * FMID + c;
    float v = agg[i] / (den[(size_t)n * H1N + (c >> 6)] + SM_EPS) + b1[c];
    agg[i] = v;
    sum += v; sq += v * v;
  }
  part[(size_t)bb * FMID + c] = sum;
  part[(size_t)(PB + bb) * FMID + c] = sq;
}

__global__ void bnreduce(const float* __restrict__ part, const float* __restrict__ gamma,
                         const float* __restrict__ beta, float* __restrict__ scale,
                         float* __restrict__ shift, int N) {
  int c = threadIdx.x;
  float sum = 0.f, sq = 0.f;
  for (int b = 0; b < PB; ++b) {
    sum += part[(size_t)b * FMID + c];
    sq  += part[(size_t)(PB + b) * FMID + c];
  }
  float mu  = sum / (float)N;
  float var = sq / (float)N - mu * mu;
  float sc  = gamma[c] * rsqrtf(var + BN_EPS);
  scale[c] = sc;
  shift[c] = beta[c] - mu * sc;
}

__global__ void bnapply_elu(float* __restrict__ a, const float* __restrict__ scale,
                            const float* __restrict__ shift, size_t total) {
  size_t i = (size_t)blockIdx.x * blockDim.x + threadIdx.x;
  size_t stride = (size_t)gridDim.x * blockDim.x;
  for (; i < total; i += stride) {
    int c = (int)(i & (FMID - 1));
    float v = a[i] * scale[c] + shift[c];
    a[i] = v > 0.f ? v : expm1f(v);
  }
}

// --------------------------------------------- layer2 scores (H=1; h2 padded 48)
__global__ void scores2(const float* __restrict__ h2, const float* __restrict__ asrc,
                        const float* __restrict__ adst, float* __restrict__ s,
                        float* __restrict__ d, int N) {
  int n = blockIdx.x * blockDim.x + threadIdx.x;
  if (n >= N) return;
  const float* hp = h2 + (size_t)n * NCP;
  float ss = 0.f, dd = 0.f;
  for (int c = 0; c < NC; ++c) { float v = hp[c]; ss += v * asrc[c]; dd += v * adst[c]; }
  s[n] = ss; d[n] = dd;
}

__global__ void edgeA2(const long long* __restrict__ ei, const float* __restrict__ s,
                       const float* __restrict__ d, unsigned* __restrict__ m, int E, int N) {
  int e = blockIdx.x * blockDim.x + threadIdx.x;
  if (e >= E + N) return;
  int src, dst;
  if (e < E) { src = (int)ei[e]; dst = (int)ei[(size_t)E + e]; }
  else       { src = dst = e - E; }
  atomicMax(&m[dst], encf(lrelu(s[src] + d[dst])));
}

__global__ void edgeB2(const long long* __restrict__ ei, const float* __restrict__ s,
                       const float* __restrict__ d, const unsigned* __restrict__ m,
                       float* __restrict__ den, const float* __restrict__ h2,
                       float* __restrict__ out, int E, int N) {
  int gw = (int)((blockIdx.x * blockDim.x + threadIdx.x) >> 5);
  int lane = threadIdx.x & 31;
  if (gw >= E + N) return;
  int src, dst;
  if (gw < E) { src = (int)ei[gw]; dst = (int)ei[(size_t)E + gw]; }
  else        { src = dst = gw - E; }
  float p = expf(lrelu(s[src] + d[dst]) - decf(m[dst]));
  if (lane == 0) atomicAdd(&den[dst], p);
  const float* hs = h2 + (size_t)src * NCP;   // padded rows
  float* op = out + (size_t)dst * NC;         // output rows are tight [N,40]
  for (int ch = lane; ch < NC; ch += 32) atomicAdd(&op[ch], hs[ch] * p);
}

// ------------------- normalize, +b2, row log_softmax (one wave per node, NC=40)
__global__ void finalize(float* __restrict__ out, const float* __restrict__ den,
                         const float* __restrict__ b2, int N) {
  int gw = (int)((blockIdx.x * blockDim.x + threadIdx.x) >> 5);
  int lane = threadIdx.x & 31;
  if (gw >= N) return;
  float* op = out + (size_t)gw * NC;
  float dn = den[gw] + SM_EPS;
  float v0 = op[lane] / dn + b2[lane];                         // lanes 0..31 valid
  bool hi = lane < (NC - 32);                                  // lanes 0..7 carry 32..39
  float v1 = hi ? (op[32 + lane] / dn + b2[32 + lane]) : 0.f;
  float mx = hi ? fmaxf(v0, v1) : v0;
  for (int off = 16; off; off >>= 1) mx = fmaxf(mx, __shfl_xor(mx, off, 32));
  float sm = expf(v0 - mx) + (hi ? expf(v1 - mx) : 0.f);
  for (int off = 16; off; off >>= 1) sm += __shfl_xor(sm, off, 32);
  float lse = mx + logf(sm);
  op[lane] = v0 - lse;
  if (hi) op[32 + lane] = v1 - lse;
}

// ---------------------------------------------------------------------------
extern "C" void kernel_launch(void* const* d_in, const int* in_sizes, int n_in,
                              void* d_out, int out_size, void* d_ws, size_t ws_size,
                              hipStream_t stream) {
  const float*     x     = (const float*)d_in[0];
  const long long* ei    = (const long long*)d_in[1];   // int64 [2,E]
  const float*     W1    = (const float*)d_in[2];
  const float*     as1   = (const float*)d_in[3];
  const float*     ad1   = (const float*)d_in[4];
  const float*     b1    = (const float*)d_in[5];
  const float*     gamma = (const float*)d_in[6];
  const float*     beta  = (const float*)d_in[7];
  const float*     W2    = (const float*)d_in[8];
  const float*     as2   = (const float*)d_in[9];
  const float*     ad2   = (const float*)d_in[10];
  const float*     b2    = (const float*)d_in[11];
  float* out = (float*)d_out;

  const int N = in_sizes[0] / FIN;     // 50000 (multiple of 16)
  const int E = in_sizes[1] / 2;       // 400000
  const int EP = E + N;                // edges incl. self-loops

  // ---- workspace layout (floats) ----
  float*    h1   = (float*)d_ws;                        // [N,256]
  float*    agg1 = h1 + (size_t)N * FMID;               // [N,256]
  float*    s1   = agg1 + (size_t)N * FMID;             // [N,4]
  float*    d1v  = s1 + (size_t)N * H1N;                // [N,4]
  float*    den1 = d1v + (size_t)N * H1N;               // [N,4]   (zeroed)
  unsigned* m1   = (unsigned*)(den1 + (size_t)N * H1N); // [N,4]   (zeroed)
  float*    h2   = (float*)(m1 + (size_t)N * H1N);      // [N,48] padded
  float*    s2   = h2 + (size_t)N * NCP;                // [N]
  float*    d2v  = s2 + N;                              // [N]
  float*    den2 = d2v + N;                             // [N]     (zeroed)
  unsigned* m2   = (unsigned*)(den2 + N);               // [N]     (zeroed)
  float*    w1p  = (float*)(m2 + N);                    // [256*256] packed W1
  float*    w2p  = w1p + (size_t)FIN * FMID;            // [256*48] packed+padded W2
  float*    part = w2p + (size_t)FMID * NCP;            // [2*PB,256]
  float*    scale = part + (size_t)2 * PB * FMID;       // [256]
  float*    shift = scale + FMID;                       // [256]

  const int T = 256;
  // zero: agg1 | den1+m1 | den2+m2 | d_out ; pack weights
  {
    size_t n1 = (size_t)N * FMID;
    zero_u32<<<(int)((n1 + T - 1) / T), T, 0, stream>>>((unsigned*)agg1, n1);
    size_t n2 = (size_t)N * H1N * 2;   // den1 + m1 contiguous
    zero_u32<<<(int)((n2 + T - 1) / T), T, 0, stream>>>((unsigned*)den1, n2);
    size_t n3 = (size_t)N * 2;         // den2 + m2 contiguous
    zero_u32<<<(int)((n3 + T - 1) / T), T, 0, stream>>>((unsigned*)den2, n3);
    size_t n4 = (size_t)N * NC;
    zero_u32<<<(int)((n4 + T - 1) / T), T, 0, stream>>>((unsigned*)out, n4);
    int p1 = (FIN >> 2) * 2 * FMID;    // W1 pair count
    packW<<<(p1 + T - 1) / T, T, 0, stream>>>(W1, w1p, FIN, FMID, FMID);
    int p2 = (FMID >> 2) * 2 * NCP;    // W2 pair count (zero-pads cols 40..47)
    packW<<<(p2 + T - 1) / T, T, 0, stream>>>(W2, w2p, FMID, NCP, NC);
  }

  // ---- layer 1 ----
  {
    int rt = N / 16, cg = FMID / 64;                   // CT=4 -> 4 col groups
    int waves = rt * cg;                               // 12500 waves
    gemm_f32_wmma<4><<<(waves + 7) / 8, T, 0, stream>>>(x, w1p, h1, rt, cg, FIN, FMID);
  }
  scores1<<<N, T, 0, stream>>>(h1, as1, ad1, s1, d1v, N);
  edgeA1<<<(EP * H1N + T - 1) / T, T, 0, stream>>>(ei, s1, d1v, m1, E, N);
  edgeB1<<<(EP + 7) / 8, T, 0, stream>>>(ei, s1, d1v, m1, den1, h1, agg1, E, N);
  bnstats<<<PB, T, 0, stream>>>(agg1, den1, b1, part, N);
  bnreduce<<<1, T, 0, stream>>>(part, gamma, beta, scale, shift, N);
  bnapply_elu<<<4096, T, 0, stream>>>(agg1, scale, shift, (size_t)N * FMID);

  // ---- layer 2 ----
  {
    int rt = N / 16, cg = 1;                           // CT=3 covers all 48 cols
    int waves = rt * cg;                               // 3125 waves
    gemm_f32_wmma<3><<<(waves + 7) / 8, T, 0, stream>>>(agg1, w2p, h2, rt, cg, FMID, NCP);
  }
  scores2<<<(N + T - 1) / T, T, 0, stream>>>(h2, as2, ad2, s2, d2v, N);
  edgeA2<<<(EP + T - 1) / T, T, 0, stream>>>(ei, s2, d2v, m2, E, N);
  edgeB2<<<(EP + 7) / 8, T, 0, stream>>>(ei, s2, d2v, m2, den2, h2, out, E, N);
  finalize<<<(N + 7) / 8, T, 0, stream>>>(out, den2, b2, N);
}